// ReconGraph_50611894616772
// MI455X (gfx1250) — compile-verified
//
#include <hip/hip_runtime.h>
#include <cfloat>
#include <cstdint>

#define NDIM 8192
#define TILE 64
#define SIN_STRIDE 68                  // floats per LDS row: 272B, 16B-aligned
#define SIN_ROWS   66                  // rows i0-1 .. i0+64
#define SOUT_STRIDE (TILE + 1)         // 65 -> conflict-free transposed writes
#define NTHREADS 256

// In-row layout of s_in (per row r_s, global row = i0 + r_s - 1):
//   idx 0..63 : global cols j0 .. j0+63   (interior, 16B-aligned)
//   idx 64    : global col  j0+64         (east halo; natural c+1 for c=63)
//   idx 66    : global col  j0-1          (west halo)

#if __has_builtin(__builtin_amdgcn_global_load_async_to_lds_b32)
#define HAVE_ASYNC32 1
#else
#define HAVE_ASYNC32 0
#endif
#if __has_builtin(__builtin_amdgcn_global_load_async_to_lds_b128)
#define HAVE_ASYNC128 1
#else
#define HAVE_ASYNC128 0
#endif

typedef int v4i __attribute__((ext_vector_type(4)));
typedef __attribute__((address_space(1))) int g_int_t;   // global (AS1) scalar
typedef __attribute__((address_space(3))) int l_int_t;   // LDS (AS3) scalar
typedef __attribute__((address_space(1))) v4i g_v4i_t;   // global (AS1) vec4
typedef __attribute__((address_space(3))) v4i l_v4i_t;   // LDS (AS3) vec4

__global__ __launch_bounds__(NTHREADS) void recon_connect_kernel(
    const float* __restrict__ d, const float* __restrict__ thr_p,
    float* __restrict__ out)
{
    __shared__ float s_in[SIN_ROWS * SIN_STRIDE];
    __shared__ float s_out[TILE * SOUT_STRIDE];

    const int tid = threadIdx.x;
    const int i0  = blockIdx.y * TILE;   // input row base
    const int j0  = blockIdx.x * TILE;   // input col base
    const float thr = thr_p[0];

    // ---- Phase 1a: bulk 66 rows x 64 interior cols via async b128 DMA ----
    for (int k = tid; k < SIN_ROWS * 16; k += NTHREADS) {
        int r_s = k >> 4;                 // 0..65
        int kc  = k & 15;                 // 16B chunk within row
        int gi  = i0 + r_s - 1;
        float* dst = &s_in[r_s * SIN_STRIDE + (kc << 2)];
        if ((unsigned)gi < (unsigned)NDIM) {
#if HAVE_ASYNC128
            __builtin_amdgcn_global_load_async_to_lds_b128(
                (g_v4i_t*)(d + (size_t)gi * NDIM + j0 + (kc << 2)),
                (l_v4i_t*)dst, /*offset=*/0, /*cpol=*/0);
#else
            const float4 v = *(const float4*)(d + (size_t)gi * NDIM + j0 + (kc << 2));
            *(float4*)dst = v;
#endif
        } else {
            *(float4*)dst = make_float4(FLT_MAX, FLT_MAX, FLT_MAX, FLT_MAX);
        }
    }

    // ---- Phase 1b: west/east halo columns (b32 async DMA) ----
    if (tid < SIN_ROWS * 2) {
        int r_s  = tid >> 1;
        int east = tid & 1;
        int gi   = i0 + r_s - 1;
        int gj   = east ? (j0 + TILE) : (j0 - 1);
        int dstc = east ? TILE : 66;
        float* dst = &s_in[r_s * SIN_STRIDE + dstc];
        bool inb = ((unsigned)gi < (unsigned)NDIM) && ((unsigned)gj < (unsigned)NDIM);
        if (inb) {
#if HAVE_ASYNC32
            __builtin_amdgcn_global_load_async_to_lds_b32(
                (g_int_t*)(d + (size_t)gi * NDIM + gj),
                (l_int_t*)dst, /*offset=*/0, /*cpol=*/0);
#else
            *dst = d[(size_t)gi * NDIM + gj];
#endif
        } else {
            *dst = FLT_MAX;               // border padding, same as jnp.pad(..., big)
        }
    }

#if HAVE_ASYNC32 || HAVE_ASYNC128
#if __has_builtin(__builtin_amdgcn_s_wait_asynccnt)
    __builtin_amdgcn_s_wait_asynccnt(0);
#else
    asm volatile("s_wait_asynccnt 0x0" ::: "memory");
#endif
#endif
    __syncthreads();

    // ---- Phase 2: diagonal connectivity, rolling registers down a column ----
    {
        const int c  = tid & (TILE - 1);        // owned column, 0..63
        const int rb = (tid >> 6) << 4;         // row block base: 0,16,32,48
        const int cw = (c == 0) ? 66 : (c - 1); // west neighbor in-row index
        const int ce = c + 1;                   // east neighbor (63 -> 64 halo)

        const float* rp = &s_in[rb * SIN_STRIDE];
        float w0 = rp[cw];                  // s_in row rb
        float e0 = rp[ce];
        float w1 = rp[SIN_STRIDE + cw];     // s_in row rb+1
        float e1 = rp[SIN_STRIDE + ce];
#pragma unroll
        for (int rr = 0; rr < 16; ++rr) {
            const float* bp = &s_in[(rb + rr) * SIN_STRIDE];
            float cen = bp[SIN_STRIDE + c];       // center: s_in row r+1, col c
            float w2  = bp[2 * SIN_STRIDE + cw];  // bottom-west: s_in row r+2
            float e2  = bp[2 * SIN_STRIDE + ce];  // bottom-east
            bool conn = (__builtin_fabsf(w0 - cen) <= thr) |
                        (__builtin_fabsf(e0 - cen) <= thr) |
                        (__builtin_fabsf(w2 - cen) <= thr) |
                        (__builtin_fabsf(e2 - cen) <= thr);
            s_out[c * SOUT_STRIDE + (rb + rr)] = conn ? 1.0f : 0.0f;  // transposed
            w0 = w1; w1 = w2;
            e0 = e1; e1 = e2;
        }
    }
    __syncthreads();

    // ---- Phase 3: coalesced float4 stores of the transposed tile ----
    // out has shape (N, M): out[j, i] = connected[i, j]
    for (int k = tid; k < (TILE * TILE) / 4; k += NTHREADS) {
        int r  = k >> 4;          // output row   (j - j0), 0..63
        int c4 = (k & 15) << 2;   // output col4  (i - i0), multiple of 4
        const float* s = &s_out[r * SOUT_STRIDE + c4];
        float4 v = make_float4(s[0], s[1], s[2], s[3]);
        *(float4*)(out + (size_t)(j0 + r) * NDIM + (size_t)(i0 + c4)) = v;
    }
}

extern "C" void kernel_launch(void* const* d_in, const int* in_sizes, int n_in,
                              void* d_out, int out_size, void* d_ws, size_t ws_size,
                              hipStream_t stream) {
    const float* d_noised = (const float*)d_in[0];
    const float* thr      = (const float*)d_in[1];   // single-element array
    float* out            = (float*)d_out;

    dim3 grid(NDIM / TILE, NDIM / TILE);   // 128 x 128 blocks
    recon_connect_kernel<<<grid, NTHREADS, 0, stream>>>(d_noised, thr, out);
}